// randomized_encoding_43774306681242
// MI455X (gfx1250) — compile-verified
//
#include <hip/hip_runtime.h>

// Problem dims (fixed by the reference)
#define NB   32      // batch
#define NC   1024    // channels == P
#define NPIX 784     // H*W
#define NM   4       // ELM models
#define EPSV 1e-10f

typedef __attribute__((ext_vector_type(2))) float v2f;
typedef __attribute__((ext_vector_type(8))) float v8f;

__device__ __forceinline__ float wave_sum32(float v) {
    v += __shfl_xor(v, 16, 32);
    v += __shfl_xor(v, 8, 32);
    v += __shfl_xor(v, 4, 32);
    v += __shfl_xor(v, 2, 32);
    v += __shfl_xor(v, 1, 32);
    return v;
}

__device__ __forceinline__ float nan_fix(float v) {
    const float FMAX = 3.4028234663852886e38f;
    if (v != v) return 0.0f;         // NaN -> 0
    if (v > FMAX) return FMAX;       // +inf -> max
    if (v < -FMAX) return -FMAX;     // -inf -> min
    return v;
}

// ---------------------------------------------------------------------------
// Kernel 1: per-(b,c) reciprocal L2 norm over the 784 pixels.
// 8 waves per block, one wave per channel.
// ---------------------------------------------------------------------------
__global__ void k_norms(const float* __restrict__ x, float* __restrict__ rnorm) {
    const int wave = threadIdx.x >> 5;
    const int lane = threadIdx.x & 31;
    const int bc = blockIdx.x * 8 + wave;          // 0 .. NB*NC-1
    const float* px = x + (size_t)bc * NPIX;
    float s = 0.0f;
    for (int i = lane; i < NPIX; i += 32) {
        float v = px[i];
        s += v * v;
    }
    s = wave_sum32(s);
    if (lane == 0) rnorm[bc] = 1.0f / fmaxf(sqrtf(s), EPSV);
}

// ---------------------------------------------------------------------------
// Kernel 2: G = sigmoid( (W * rn) @ X_raw )  via V_WMMA_F32_16X16X4_F32.
// A (16x4 tiles): rows = 16 padded model rows (4 valid), K = channels.
// B (4x16 tiles): K = channels, cols = 16 pixels.
// One wave per block; block = (pixel tile n0, batch b).
// ---------------------------------------------------------------------------
__global__ void __launch_bounds__(32)
k_gmat(const float* __restrict__ x, const float* __restrict__ Wm,
       const float* __restrict__ rnorm, float* __restrict__ g) {
    __shared__ float Wrn[NM * NC];                 // 16 KB: W[m,c] * rn[b,c]
    const int b = blockIdx.y;
    const int n0 = blockIdx.x * 16;
    const int tid = threadIdx.x;

    for (int i = tid; i < NM * NC; i += 32) {
        const int c = i & (NC - 1);
        Wrn[i] = Wm[i] * rnorm[b * NC + c];
    }
    __syncthreads();

    const int half = tid >> 4;                     // 0: lanes 0-15, 1: lanes 16-31
    const int lm = tid & 15;
    const float amask = (lm < NM) ? 1.0f : 0.0f;   // pad model rows 4..15 with 0
    const int mrow = lm & (NM - 1);
    const float* xb = x + (size_t)b * NC * NPIX;

    v8f acc = {};
    for (int k0 = 0; k0 < NC; k0 += 4) {
        const int c0 = k0 + 2 * half;              // K in {2h, 2h+1} per fp32 layout
        v2f a, bf;
        a[0] = Wrn[mrow * NC + c0] * amask;
        a[1] = Wrn[mrow * NC + c0 + 1] * amask;
        bf[0] = xb[c0 * NPIX + n0 + lm];
        bf[1] = xb[(c0 + 1) * NPIX + n0 + lm];
        acc = __builtin_amdgcn_wmma_f32_16x16x4_f32(
            false, a, false, bf, (short)0, acc, false, false);
    }

    // D layout: VGPR r holds row (r + 8*half); rows 0..3 are the models.
    if (half == 0) {
#pragma unroll
        for (int m = 0; m < NM; ++m) {
            const float z = acc[m];
            const float gv = 1.0f / (1.0f + __expf(-z));
            g[(b * NM + m) * NPIX + n0 + lm] = gv;
        }
    }
}

// ---------------------------------------------------------------------------
// Kernel 3: denom[b,m] = sum_n g[b,m,n]^2.  One wave per (b,m).
// ---------------------------------------------------------------------------
__global__ void k_denom(const float* __restrict__ g, float* __restrict__ denom) {
    const int lane = threadIdx.x & 31;
    const int bm = blockIdx.x;                     // 0 .. NB*NM-1
    const float* pg = g + bm * NPIX;
    float s = 0.0f;
    for (int i = lane; i < NPIX; i += 32) {
        float v = pg[i];
        s += v * v;
    }
    s = wave_sum32(s);
    if (lane == 0) denom[bm] = s;
}

// ---------------------------------------------------------------------------
// Kernel 4: out[b,p] = nan_fix( rn[b,p] * sum_m dot(x[b,p,:], g[b,m,:]) / denom[b,m] )
// A (16x4 tiles): rows = 16 channel rows, K = pixels.
// B (4x16 tiles): K = pixels, cols = 16 padded model cols (4 valid).
// One wave per block; block = (channel tile p0, batch b).
// ---------------------------------------------------------------------------
__global__ void __launch_bounds__(32)
k_beta(const float* __restrict__ x, const float* __restrict__ g,
       const float* __restrict__ denom, const float* __restrict__ rnorm,
       float* __restrict__ out) {
    __shared__ float gl[NM * NPIX];                // 12.5 KB: g for this sample
    const int b = blockIdx.y;
    const int p0 = blockIdx.x * 16;
    const int tid = threadIdx.x;

    for (int i = tid; i < NM * NPIX; i += 32) gl[i] = g[b * NM * NPIX + i];
    __syncthreads();

    const int half = tid >> 4;
    const int lm = tid & 15;
    const float bmask = (lm < NM) ? 1.0f : 0.0f;   // pad model cols 4..15 with 0
    const int mcol = lm & (NM - 1);
    const float* xb = x + (size_t)b * NC * NPIX;

    v8f acc = {};
    for (int k0 = 0; k0 < NPIX; k0 += 4) {         // 196 iterations
        const int kk = k0 + 2 * half;
        v2f a, bf;
        a[0] = xb[(p0 + lm) * NPIX + kk];
        a[1] = xb[(p0 + lm) * NPIX + kk + 1];
        bf[0] = gl[mcol * NPIX + kk] * bmask;
        bf[1] = gl[mcol * NPIX + kk + 1] * bmask;
        acc = __builtin_amdgcn_wmma_f32_16x16x4_f32(
            false, a, false, bf, (short)0, acc, false, false);
    }

    // Per-column scale by 1/denom, then butterfly-sum the 4 model columns.
    const float rd0 = 1.0f / denom[b * NM + 0];
    const float rd1 = 1.0f / denom[b * NM + 1];
    const float rd2 = 1.0f / denom[b * NM + 2];
    const float rd3 = 1.0f / denom[b * NM + 3];
    const float q = (lm == 0) ? rd0 : (lm == 1) ? rd1
                   : (lm == 2) ? rd2 : (lm == 3) ? rd3 : 0.0f;

    float t[8];
#pragma unroll
    for (int r = 0; r < 8; ++r) {
        float v = acc[r] * q;                      // zero outside lm<4
        v += __shfl_xor(v, 1, 32);                 // sum within groups of 4 lanes
        v += __shfl_xor(v, 2, 32);
        t[r] = v;                                  // lane lm==0 of each half: full sum
    }

    if (lm == 0) {
#pragma unroll
        for (int r = 0; r < 8; ++r) {
            const int p = p0 + r + 8 * half;       // D row = r + 8*half
            out[b * NC + p] = nan_fix(t[r] * rnorm[b * NC + p]);
        }
    }
}

// ---------------------------------------------------------------------------
extern "C" void kernel_launch(void* const* d_in, const int* in_sizes, int n_in,
                              void* d_out, int out_size, void* d_ws, size_t ws_size,
                              hipStream_t stream) {
    const float* x = (const float*)d_in[0];    // [B, C, H, W] fp32
    const float* Wm = (const float*)d_in[1];   // [M, Q, P]  fp32
    float* out = (float*)d_out;                // [B, Q*P]   fp32
    float* ws = (float*)d_ws;

    float* rnorm = ws;                         // NB*NC       = 32768 floats
    float* g = rnorm + NB * NC;                // NB*NM*NPIX  = 100352 floats
    float* denom = g + NB * NM * NPIX;         // NB*NM       = 128 floats

    (void)in_sizes; (void)n_in; (void)out_size; (void)ws_size;

    k_norms<<<dim3(NB * NC / 8), dim3(256), 0, stream>>>(x, rnorm);
    k_gmat<<<dim3(NPIX / 16, NB), dim3(32), 0, stream>>>(x, Wm, rnorm, g);
    k_denom<<<dim3(NB * NM), dim3(32), 0, stream>>>(g, denom);
    k_beta<<<dim3(NC / 16, NB), dim3(32), 0, stream>>>(x, g, denom, rnorm, out);
}